// NerfRenderer_51049981280729
// MI455X (gfx1250) — compile-verified
//
#include <hip/hip_runtime.h>

// ---------------------------------------------------------------------------
// NeRF-style ray marcher for gfx1250 (MI455X), wave32.
//   - 1 wave per ray, 32 samples per loop iteration (lane == sample-in-chunk)
//   - MLP heads via V_WMMA_F32_16X16X4_F32 in transposed form (weights are
//     per-wave register-resident A fragments; per-sample masking lane-local)
//   - Cross-lane traffic via v_permlane16/v_permlanex16 (VALU, no LDS pipe).
//     All permlane calls are UNCONDITIONAL statements + cndmask blends so the
//     convergent builtins never end up inside EXEC-masked diamonds.
//   - transmittance: permlane scan + readlane carry, uniform early exit.
// ---------------------------------------------------------------------------

typedef __attribute__((ext_vector_type(2))) float v2f;
typedef __attribute__((ext_vector_type(8))) float v8f;

#define GRID_N     128
#define FEAT_DIM   16
#define DELTA      0.0016914558667664817f   /* sqrt(3)/1024 */
#define EARLY_TERM 1.0e-4f

static __device__ __forceinline__ v8f wmma4(v2f a, v2f b, v8f c) {
  // D = A(16x4 f32) * B(4x16 f32) + C(16x16 f32)
  return __builtin_amdgcn_wmma_f32_16x16x4_f32(
      false, a, false, b, (short)0, c, false, false);
}

// ---- permlane helpers ------------------------------------------------------
static __device__ __forceinline__ float pl16(float v, unsigned lo, unsigned hi) {
  return __int_as_float(__builtin_amdgcn_permlane16(
      __float_as_int(v), __float_as_int(v), lo, hi, false, false));
}
static __device__ __forceinline__ float plx16(float v, unsigned lo, unsigned hi) {
  return __int_as_float(__builtin_amdgcn_permlanex16(
      __float_as_int(v), __float_as_int(v), lo, hi, false, false));
}
// every lane <- other half's lane (laneid & 15)   (involution: swap halves)
static __device__ __forceinline__ float xhalf(float v) {
  return plx16(v, 0x76543210u, 0xfedcba98u);
}
// butterfly sum over all 32 lanes (xor offsets 1,2,4,8 in-half, 16 cross)
static __device__ __forceinline__ float wave_sum(float a) {
  a += pl16(a, 0x67452301u, 0xefcdab89u);  // ^1
  a += pl16(a, 0x45670123u, 0xcdef89abu);  // ^2
  a += pl16(a, 0x01234567u, 0x89abcdefu);  // ^4
  a += pl16(a, 0xfedcba98u, 0x76543210u);  // ^8
  a += xhalf(a);                           // ^16
  return a;
}

static __device__ __forceinline__ float softplusf(float x) {
  float e = __expf(x);
  float r = __logf(1.0f + e);
  return (x > 20.0f) ? x : r;
}
static __device__ __forceinline__ float sigmoidf(float x) {
  return 1.0f / (1.0f + __expf(-x));
}

// Combined head weights: row 0 = sigma, rows 1..3 = rgb, rows 4..15 = 0.
// Column k: k<16 -> feat_k, k=16..18 -> dir_xyz, k=19 -> pad.
static __device__ __forceinline__ float w2big(int r, int k,
                                              const float* __restrict__ ws,
                                              const float* __restrict__ wr) {
  if (r == 0) return (k < 16) ? ws[k] : 0.0f;
  if (r <= 3) return (k < 19) ? wr[k * 3 + (r - 1)] : 0.0f;
  return 0.0f;
}

// Decode one 16-sample group:
//  D1 = A1(w_feat^T 16x4) * B1(samples^T 4x16) + bias  -> feat^T, relu, mask
//  D2 = A2(W2 16x20) * [feat;dirs]^T(20x16) + bias2    -> (sigma,r,g,b) rows
static __device__ __forceinline__ v8f decode_group(
    v2f a1, v2f b1, v8f c1, const float (&a2)[5][2], v8f c2init,
    float keep, float db0, float db1, bool hi) {
  v8f d1 = wmma4(a1, b1, c1);
#pragma unroll
  for (int r = 0; r < 8; ++r) d1[r] = keep * fmaxf(d1[r], 0.0f);

  v8f c2 = c2init;
#pragma unroll
  for (int q = 0; q < 4; ++q) {
    // feat^T element (k, n=m) lives in d1[k&7]: lanes 0-15 hold k<8,
    // lanes 16-31 hold k>=8.  B fragment: vgpr0 K=4q(+2 on hi lanes).
    float b20, b21;
    if (4 * q < 8) {  // source rows live in the LOW half
      float xa = xhalf(d1[(4 * q + 2) & 7]);   // unconditional (convergent)
      float xb = xhalf(d1[(4 * q + 3) & 7]);
      b20 = hi ? xa : d1[(4 * q + 0) & 7];
      b21 = hi ? xb : d1[(4 * q + 1) & 7];
    } else {          // source rows live in the HIGH half
      float xa = xhalf(d1[(4 * q + 0) & 7]);
      float xb = xhalf(d1[(4 * q + 1) & 7]);
      b20 = hi ? d1[(4 * q + 2) & 7] : xa;
      b21 = hi ? d1[(4 * q + 3) & 7] : xb;
    }
    v2f a = {a2[q][0], a2[q][1]};
    v2f b = {b20, b21};
    c2 = wmma4(a, b, c2);
  }
  // K-chunk 4: ray direction rows (16..19), uniform per wave.
  v2f a = {a2[4][0], a2[4][1]};
  v2f b = {db0, db1};
  return wmma4(a, b, c2);
}

__global__ __launch_bounds__(256) void nerf_march_wmma(
    const float* __restrict__ rays_o, const float* __restrict__ rays_d,
    const float* __restrict__ grid, const float* __restrict__ w_feat,
    const float* __restrict__ b_feat, const float* __restrict__ w_sigma,
    const float* __restrict__ b_sigma, const float* __restrict__ w_rgb,
    const float* __restrict__ b_rgb, float* __restrict__ out, int n_rays) {
  const int lane = threadIdx.x & 31;
  const int ray = (blockIdx.x * blockDim.x + threadIdx.x) >> 5;
  if (ray >= n_rays) return;  // wave-uniform: EXEC stays all-1s for WMMA
  const int m = lane & 15;
  const bool hi = lane >= 16;

  const float ox = rays_o[ray * 3 + 0], oy = rays_o[ray * 3 + 1],
              oz = rays_o[ray * 3 + 2];
  const float dx = rays_d[ray * 3 + 0], dy = rays_d[ray * 3 + 1],
              dz = rays_d[ray * 3 + 2];

  // --- per-wave constant WMMA fragments (loaded once) --------------------
  // A1 = w_feat^T (16x4, K-pad): lane m is feat row m; vgpr0 K = hi?2:0.
  v2f a1;
  {
    float w0 = w_feat[0 * FEAT_DIM + m];
    float w1 = w_feat[1 * FEAT_DIM + m];
    float w2 = w_feat[2 * FEAT_DIM + m];
    a1[0] = hi ? w2 : w0;
    a1[1] = hi ? 0.0f : w1;
  }
  // C1 bias: D1 row = featdim = r + 8*hi.
  v8f c1;
#pragma unroll
  for (int r = 0; r < 8; ++r) {
    float blo = b_feat[r];
    float bhi = b_feat[r + 8];
    c1[r] = hi ? bhi : blo;
  }
  // A2 chunks: W2big rows, lane m = row m; chunk q vgpr j -> K=4q+j(+2 if hi).
  float a2[5][2];
#pragma unroll
  for (int q = 0; q < 5; ++q) {
    int k0 = 4 * q + (hi ? 2 : 0);
    a2[q][0] = w2big(m, k0, w_sigma, w_rgb);
    a2[q][1] = w2big(m, k0 + 1, w_sigma, w_rgb);
  }
  // C2 bias: row0=sigma, rows1..3=rgb (rows 8..15 on hi lanes are padding).
  v8f c2init;
  {
    float bs = b_sigma[0];
    float br0 = b_rgb[0], br1 = b_rgb[1], br2 = b_rgb[2];
    c2init[0] = hi ? 0.0f : bs;
    c2init[1] = hi ? 0.0f : br0;
    c2init[2] = hi ? 0.0f : br1;
    c2init[3] = hi ? 0.0f : br2;
    c2init[4] = 0.0f;
    c2init[5] = 0.0f;
    c2init[6] = 0.0f;
    c2init[7] = 0.0f;
  }
  // direction K-chunk (rows 16..19 = dx,dy,dz,0), same for both groups
  const float db0 = hi ? dz : dx;
  const float db1 = hi ? 0.0f : dy;

  float acc0 = 0.0f, acc1 = 0.0f, acc2 = 0.0f;
  float T = 1.0f;  // carried transmittance (wave-uniform value)

  for (int chunk = 0; chunk < 32; ++chunk) {
    if (T <= EARLY_TERM) break;  // all later weights <= T < thresh

    // ---- sample position + mip360 contraction (lane = sample) ----------
    const float t = (float)(chunk * 32 + lane) * DELTA;
    float px = fmaf(dx, t, ox), py = fmaf(dy, t, oy), pz = fmaf(dz, t, oz);
    float n = fmaxf(fabsf(px), fmaxf(fabsf(py), fabsf(pz)));
    float inv = 1.0f / n;
    float sc = (n <= 1.0f) ? 0.5f : ((2.0f - inv) * inv * 0.5f);
    float cx = px * sc, cy = py * sc, cz = pz * sc;

    // ---- trilinear occupancy (grid_sample, align_corners=False) --------
    float ixf = fmaf(cx, 64.0f, 63.5f);
    float iyf = fmaf(cy, 64.0f, 63.5f);
    float izf = fmaf(cz, 64.0f, 63.5f);
    float x0f = floorf(ixf), y0f = floorf(iyf), z0f = floorf(izf);
    float wx = ixf - x0f, wy = iyf - y0f, wz = izf - z0f;
    int x0 = (int)x0f, y0 = (int)y0f, z0 = (int)z0f;
    auto tap = [&](int zi, int yi, int xi) -> float {
      bool v = ((unsigned)zi < (unsigned)GRID_N) &
               ((unsigned)yi < (unsigned)GRID_N) &
               ((unsigned)xi < (unsigned)GRID_N);
      int idx = ((((zi & 127) << 7) | (yi & 127)) << 7) | (xi & 127);
      float g = grid[idx];
      return v ? g : 0.0f;
    };
    float occ =
        tap(z0, y0, x0) * (1.f - wz) * (1.f - wy) * (1.f - wx) +
        tap(z0, y0, x0 + 1) * (1.f - wz) * (1.f - wy) * wx +
        tap(z0, y0 + 1, x0) * (1.f - wz) * wy * (1.f - wx) +
        tap(z0, y0 + 1, x0 + 1) * (1.f - wz) * wy * wx +
        tap(z0 + 1, y0, x0) * wz * (1.f - wy) * (1.f - wx) +
        tap(z0 + 1, y0, x0 + 1) * wz * (1.f - wy) * wx +
        tap(z0 + 1, y0 + 1, x0) * wz * wy * (1.f - wx) +
        tap(z0 + 1, y0 + 1, x0 + 1) * wz * wy * wx;
    const bool maskb = occ > 0.0f;
    const float mf = maskb ? 1.0f : 0.0f;

    // column masks: half-swap is an involution, one permlane serves both.
    const float mx = xhalf(mf);
    const float keepA = hi ? mx : mf;
    const float keepB = hi ? mf : mx;

    // ---- B1 fragments (samples^T 4x16) for the two groups ---------------
    // vgpr0: K = x (lo lanes) / z (hi lanes); vgpr1: K = y (lo) / pad (hi)
    const float cxx = xhalf(cx);
    const float cyx = xhalf(cy);
    const float czx = xhalf(cz);
    v2f b1A, b1B;
    b1A[0] = hi ? czx : cx;
    b1A[1] = hi ? 0.0f : cy;
    b1B[0] = hi ? cz : cxx;
    b1B[1] = hi ? 0.0f : cyx;

    v8f d2A = decode_group(a1, b1A, c1, a2, c2init, keepA, db0, db1, hi);
    v8f d2B = decode_group(a1, b1B, c1, a2, c2init, keepB, db0, db1, hi);

    // ---- gather this lane's (sigma, rgb) pre-activations ----------------
    // D2 row r sits in d2[r] lanes 0-15, col = sample index within group.
    const float xs = xhalf(d2B[0]);
    const float xr = xhalf(d2B[1]);
    const float xg = xhalf(d2B[2]);
    const float xb = xhalf(d2B[3]);
    float sp = hi ? xs : d2A[0];
    float rp = hi ? xr : d2A[1];
    float gp = hi ? xg : d2A[2];
    float bp = hi ? xb : d2A[3];

    float sigma = maskb ? softplusf(sp) : 0.0f;
    float la = -sigma * DELTA;

    // ---- 32-lane inclusive scan of log-attenuation (permlane) -----------
    float incl = la;
    {
      // rotate-down-by-d selects within a 16-half: lane j reads (j-d)&15
      const unsigned RLO[4] = {0x6543210Fu, 0x543210FEu, 0x3210FEDCu,
                               0xFEDCBA98u};
      const unsigned RHI[4] = {0xEDCBA987u, 0xDCBA9876u, 0xBA987654u,
                               0x76543210u};
#pragma unroll
      for (int s = 0; s < 4; ++s) {
        const int d = 1 << s;
        float us = pl16(incl, RLO[s], RHI[s]);    // same-half lane-d
        float uc = plx16(incl, RLO[s], RHI[s]);   // cross-half lane-d
        float u = ((lane & 15) >= d) ? us : uc;
        incl += (lane >= d) ? u : 0.0f;
      }
      float u16 = xhalf(incl);                    // step d=16
      incl += hi ? u16 : 0.0f;
    }
    float excl = incl - la;
    float trans = T * __expf(excl);
    float alpha = 1.0f - __expf(la);
    float w = trans * alpha;
    float wv = (maskb && (w > EARLY_TERM)) ? w : 0.0f;

    acc0 = fmaf(wv, sigmoidf(rp), acc0);
    acc1 = fmaf(wv, sigmoidf(gp), acc1);
    acc2 = fmaf(wv, sigmoidf(bp), acc2);

    // carry total attenuation (scalar broadcast via readlane)
    float tot = __int_as_float(__builtin_amdgcn_readlane(
        __float_as_int(incl), 31));
    T *= __expf(tot);
  }

  // ---- wave reduction (each lane held a distinct sample) ----------------
  acc0 = wave_sum(acc0);
  acc1 = wave_sum(acc1);
  acc2 = wave_sum(acc2);
  if (lane == 0) {
    out[ray * 3 + 0] = acc0;
    out[ray * 3 + 1] = acc1;
    out[ray * 3 + 2] = acc2;
  }
}

extern "C" void kernel_launch(void* const* d_in, const int* in_sizes, int n_in,
                              void* d_out, int out_size, void* d_ws,
                              size_t ws_size, hipStream_t stream) {
  const float* rays_o = (const float*)d_in[0];
  const float* rays_d = (const float*)d_in[1];
  const float* grid = (const float*)d_in[2];
  const float* w_feat = (const float*)d_in[3];
  const float* b_feat = (const float*)d_in[4];
  const float* w_sigma = (const float*)d_in[5];
  const float* b_sigma = (const float*)d_in[6];
  const float* w_rgb = (const float*)d_in[7];
  const float* b_rgb = (const float*)d_in[8];
  float* out = (float*)d_out;

  const int n_rays = in_sizes[0] / 3;          // 8192
  const int threads = 256;                     // 8 wave32 per block
  const int blocks = (n_rays * 32 + threads - 1) / threads;
  hipLaunchKernelGGL(nerf_march_wmma, dim3(blocks), dim3(threads), 0, stream,
                     rays_o, rays_d, grid, w_feat, b_feat, w_sigma, b_sigma,
                     w_rgb, b_rgb, out, n_rays);
}